// GCN_61194694033576
// MI455X (gfx1250) — compile-verified
//
#include <hip/hip_runtime.h>
#include <hip/hip_bf16.h>

// ---------------------------------------------------------------------------
// GCN forward for MI455X (gfx1250, wave32).
//   phase 1: deg/dis                            (tiny)
//   phase 2: h0 = x @ W1 via V_WMMA_F32_16X16X4_F32, fused self-loop init
//   phase 3: edge gather/scatter (float4 + global_atomic_add_f32)
//   phase 4: bias + relu
//   phase 5: t2 = h @ W2 (VALU, bandwidth bound), fused self-loop init
//   phase 6: edge scatter layer 2
//   phase 7: bias + log_softmax
// ---------------------------------------------------------------------------

typedef float v2f __attribute__((ext_vector_type(2)));
typedef float v8f __attribute__((ext_vector_type(8)));

#define F_IN 32
#define HID  32

// ---- phase 1 --------------------------------------------------------------
__global__ void k_init_deg(float* __restrict__ deg, int n) {
    int i = blockIdx.x * blockDim.x + threadIdx.x;
    if (i < n) deg[i] = 1.0f;                       // self-loop contributes 1
}

__global__ void k_deg_edges(const int* __restrict__ ei, float* __restrict__ deg,
                            int n_edges) {
    int e = blockIdx.x * blockDim.x + threadIdx.x;
    if (e < n_edges) atomicAdd(&deg[ei[n_edges + e]], 1.0f);   // dst row
}

__global__ void k_rsqrt(float* __restrict__ deg, int n) {
    int i = blockIdx.x * blockDim.x + threadIdx.x;
    if (i < n) deg[i] = rsqrtf(deg[i]);             // deg >= 1 always
}

// ---- phase 2: WMMA fp32 GEMM  h0 = x @ W1,  agg1 = dis^2 * h0 -------------
// One wave computes rows [tile*16, tile*16+16) x all 32 output columns.
__global__ void k_gemm1_wmma(const float* __restrict__ x,
                             const float* __restrict__ W1,
                             const float* __restrict__ dis,
                             float* __restrict__ h0,
                             float* __restrict__ agg1,
                             int n_nodes, int n_tiles) {
    __shared__ float w[F_IN * HID];                 // 4 KB of W1 in LDS
    for (int i = threadIdx.x; i < F_IN * HID; i += blockDim.x) w[i] = W1[i];
    __syncthreads();

    const int lane = threadIdx.x & 31;
    const int wave = threadIdx.x >> 5;
    const int tile = blockIdx.x * (blockDim.x >> 5) + wave;
    if (tile >= n_tiles) return;                    // wave-uniform: EXEC stays full

    const int m0      = tile << 4;
    const int n       = lane & 15;                  // output col within N-tile
    const int kb_half = (lane >> 4) << 1;           // lanes 0-15 -> K 0,1 ; 16-31 -> K 2,3
    int mrow = m0 + (lane & 15);
    if (mrow >= n_nodes) mrow = n_nodes - 1;        // clamp tail reads (stores guarded)

    v8f c0 = {};                                    // N-tile 0 (cols 0..15)
    v8f c1 = {};                                    // N-tile 1 (cols 16..31)

    #pragma unroll
    for (int kk = 0; kk < 8; ++kk) {                // K = 32 = 8 x 4
        const int kb = kk * 4 + kb_half;
        v2f a;
        a.x = x[mrow * F_IN + kb];
        a.y = x[mrow * F_IN + kb + 1];
        v2f b0, b1;
        b0.x = w[kb * HID + n];           b0.y = w[(kb + 1) * HID + n];
        b1.x = w[kb * HID + 16 + n];      b1.y = w[(kb + 1) * HID + 16 + n];
        c0 = __builtin_amdgcn_wmma_f32_16x16x4_f32(false, a, false, b0,
                                                   (short)0, c0, false, false);
        c1 = __builtin_amdgcn_wmma_f32_16x16x4_f32(false, a, false, b1,
                                                   (short)0, c1, false, false);
    }

    // D layout: VGPR v, lanes 0-15 -> M=v, lanes 16-31 -> M=v+8 ; N = lane%16
    #pragma unroll
    for (int v = 0; v < 8; ++v) {
        const int m = m0 + ((lane < 16) ? v : v + 8);
        if (m >= n_nodes) continue;
        const float d = dis[m];
        const float s = d * d;                      // self-loop norm
        const float e0 = c0[v], e1 = c1[v];
        h0[m * HID + n]        = e0;
        h0[m * HID + 16 + n]   = e1;
        agg1[m * HID + n]      = s * e0;
        agg1[m * HID + 16 + n] = s * e1;
    }
}

// ---- phase 3: layer-1 edge scatter (8 threads / edge, float4) -------------
__global__ void k_scatter1(const int* __restrict__ ei,
                           const float* __restrict__ dis,
                           const float* __restrict__ h0,
                           float* __restrict__ agg1,
                           int n_edges) {
    int tid = blockIdx.x * blockDim.x + threadIdx.x;
    int e = tid >> 3;
    int q = tid & 7;
    if (e >= n_edges) return;
    const int s = ei[e];
    const int d = ei[n_edges + e];
    const float norm = dis[s] * dis[d];
    const float4 m = ((const float4*)(h0 + (size_t)s * HID))[q];
    float* o = agg1 + (size_t)d * HID + q * 4;
    atomicAdd(o + 0, norm * m.x);
    atomicAdd(o + 1, norm * m.y);
    atomicAdd(o + 2, norm * m.z);
    atomicAdd(o + 3, norm * m.w);
}

// ---- phase 4: h = relu(agg1 + b1) -----------------------------------------
__global__ void k_bias_relu(const float* __restrict__ agg1,
                            const float* __restrict__ b1,
                            float* __restrict__ h,
                            int n_vec4) {                       // n_nodes * 8
    int i = blockIdx.x * blockDim.x + threadIdx.x;
    if (i >= n_vec4) return;
    float4 v = ((const float4*)agg1)[i];
    const int f = (i & 7) * 4;
    v.x = fmaxf(v.x + b1[f + 0], 0.0f);
    v.y = fmaxf(v.y + b1[f + 1], 0.0f);
    v.z = fmaxf(v.z + b1[f + 2], 0.0f);
    v.w = fmaxf(v.w + b1[f + 3], 0.0f);
    ((float4*)h)[i] = v;
}

// ---- phase 5: t2 = h @ W2 (32->2), agg2 = dis^2 * t2 ----------------------
__global__ void k_lin2(const float* __restrict__ h,
                       const float* __restrict__ W2,      // [32,2] row-major
                       const float* __restrict__ dis,
                       float* __restrict__ t2,
                       float* __restrict__ agg2,
                       int n_nodes) {
    int i = blockIdx.x * blockDim.x + threadIdx.x;
    if (i >= n_nodes) return;
    const float4* hr = (const float4*)(h + (size_t)i * HID);
    float a0 = 0.0f, a1 = 0.0f;
    #pragma unroll
    for (int q = 0; q < 8; ++q) {
        const float4 v = hr[q];
        const int k = q * 4;
        a0 = fmaf(v.x, W2[(k + 0) * 2 + 0], a0);
        a1 = fmaf(v.x, W2[(k + 0) * 2 + 1], a1);
        a0 = fmaf(v.y, W2[(k + 1) * 2 + 0], a0);
        a1 = fmaf(v.y, W2[(k + 1) * 2 + 1], a1);
        a0 = fmaf(v.z, W2[(k + 2) * 2 + 0], a0);
        a1 = fmaf(v.z, W2[(k + 2) * 2 + 1], a1);
        a0 = fmaf(v.w, W2[(k + 3) * 2 + 0], a0);
        a1 = fmaf(v.w, W2[(k + 3) * 2 + 1], a1);
    }
    t2[i * 2 + 0] = a0;
    t2[i * 2 + 1] = a1;
    const float d = dis[i];
    const float s = d * d;
    agg2[i * 2 + 0] = s * a0;
    agg2[i * 2 + 1] = s * a1;
}

// ---- phase 6: layer-2 edge scatter ----------------------------------------
__global__ void k_scatter2(const int* __restrict__ ei,
                           const float* __restrict__ dis,
                           const float* __restrict__ t2,
                           float* __restrict__ agg2,
                           int n_edges) {
    int e = blockIdx.x * blockDim.x + threadIdx.x;
    if (e >= n_edges) return;
    const int s = ei[e];
    const int d = ei[n_edges + e];
    const float norm = dis[s] * dis[d];
    atomicAdd(&agg2[d * 2 + 0], norm * t2[s * 2 + 0]);
    atomicAdd(&agg2[d * 2 + 1], norm * t2[s * 2 + 1]);
}

// ---- phase 7: out = log_softmax(agg2 + b2) --------------------------------
__global__ void k_logsoftmax(const float* __restrict__ agg2,
                             const float* __restrict__ b2,
                             float* __restrict__ out,
                             int n_nodes) {
    int i = blockIdx.x * blockDim.x + threadIdx.x;
    if (i >= n_nodes) return;
    const float z0 = agg2[i * 2 + 0] + b2[0];
    const float z1 = agg2[i * 2 + 1] + b2[1];
    const float m  = fmaxf(z0, z1);
    const float lse = m + logf(expf(z0 - m) + expf(z1 - m));
    out[i * 2 + 0] = z0 - lse;
    out[i * 2 + 1] = z1 - lse;
}

// ---------------------------------------------------------------------------
extern "C" void kernel_launch(void* const* d_in, const int* in_sizes, int n_in,
                              void* d_out, int out_size, void* d_ws, size_t ws_size,
                              hipStream_t stream) {
    const float* x  = (const float*)d_in[0];
    const int*   ei = (const int*)  d_in[1];     // [2, E] int32
    const float* W1 = (const float*)d_in[2];
    const float* b1 = (const float*)d_in[3];
    const float* W2 = (const float*)d_in[4];
    const float* b2 = (const float*)d_in[5];
    float* out = (float*)d_out;

    const int n_nodes = in_sizes[0] / F_IN;
    const int n_edges = in_sizes[1] / 2;

    // workspace layout (floats), buffers reused across phases
    float* ws   = (float*)d_ws;
    size_t o    = 0;
    float* dis  = ws + o;  o += (size_t)((n_nodes + 3) & ~3);          // deg -> dis
    float* h0   = ws + o;  o += (size_t)n_nodes * HID;                 // h0 -> h
    float* agg1 = ws + o;  o += (size_t)n_nodes * HID;                 // agg1 -> t2
    float* agg2 = ws + o;  o += (size_t)((2 * n_nodes + 3) & ~3);
    float* t2   = agg1;                                                // reuse after relu
    float* h    = h0;                                                  // reuse after relu
    (void)ws_size; (void)n_in; (void)out_size;

    const int TB = 256;
    const int n_tiles = (n_nodes + 15) >> 4;

    k_init_deg<<<(n_nodes + TB - 1) / TB, TB, 0, stream>>>(dis, n_nodes);
    k_deg_edges<<<(n_edges + TB - 1) / TB, TB, 0, stream>>>(ei, dis, n_edges);
    k_rsqrt<<<(n_nodes + TB - 1) / TB, TB, 0, stream>>>(dis, n_nodes);

    k_gemm1_wmma<<<(n_tiles + 7) / 8, 256, 0, stream>>>(x, W1, dis, h0, agg1,
                                                        n_nodes, n_tiles);

    const long long s1_threads = (long long)n_edges * 8;
    k_scatter1<<<(int)((s1_threads + TB - 1) / TB), TB, 0, stream>>>(ei, dis, h0,
                                                                     agg1, n_edges);

    const int n_vec4 = n_nodes * (HID / 4);
    k_bias_relu<<<(n_vec4 + TB - 1) / TB, TB, 0, stream>>>(agg1, b1, h, n_vec4);

    k_lin2<<<(n_nodes + TB - 1) / TB, TB, 0, stream>>>(h, W2, dis, t2, agg2, n_nodes);

    k_scatter2<<<(n_edges + TB - 1) / TB, TB, 0, stream>>>(ei, dis, t2, agg2, n_edges);

    k_logsoftmax<<<(n_nodes + TB - 1) / TB, TB, 0, stream>>>(agg2, b2, out, n_nodes);
}